// SoftmaxTopKRouter_8332236554938
// MI455X (gfx1250) — compile-verified
//
#include <hip/hip_runtime.h>
#include <hip/hip_bf16.h>

#define NUM_TOKENS 32768
#define DIM 2048
#define NUM_EXPERTS 8
#define TOP_K 2
#define NPAD 16            // experts padded to 16 for the WMMA N dimension

#define WAVES_PER_WG 8
#define TOKENS_PER_WAVE 16
#define TOKENS_PER_WG (WAVES_PER_WG * TOKENS_PER_WAVE)

// workspace layout (floats)
#define WS_WP 0                      // padded W: 16 x 2048
#define WS_COUNTS (NPAD * DIM)       // 8 floats
#define WS_PSUM (NPAD * DIM + 8)     // 8 floats

// output layout (floats)
#define OUT_W 0
#define OUT_I (NUM_TOKENS * TOP_K)
#define OUT_AUX (2 * NUM_TOKENS * TOP_K)
#define OUT_TPE (OUT_AUX + 1)

typedef __attribute__((ext_vector_type(2))) float v2f;
typedef __attribute__((ext_vector_type(8))) float v8f;

// ds_swizzle group-of-32 XOR patterns: offset = (xor_mask<<10) | (0<<5) | 0x1f
#define SWZ_X1  0x041f
#define SWZ_X2  0x081f
#define SWZ_X4  0x101f
#define SWZ_X16 0x401f

template <int P>
__device__ __forceinline__ float swzf(float v) {
    return __int_as_float(__builtin_amdgcn_ds_swizzle(__float_as_int(v), P));
}
template <int P>
__device__ __forceinline__ int swzi(int v) {
    return __builtin_amdgcn_ds_swizzle(v, P);
}

// one branch-free argmax butterfly step (max value, lowest index on ties)
template <int P>
__device__ __forceinline__ void amax_step(float& v, int& i) {
    float ov = swzf<P>(v);
    int   oi = swzi<P>(i);
    bool take = (ov > v) | ((ov == v) & (oi < i));
    v = take ? ov : v;
    i = take ? oi : i;
}

__device__ __forceinline__ void argmax8(float& v, int& i) {
    amax_step<SWZ_X1>(v, i);
    amax_step<SWZ_X2>(v, i);
    amax_step<SWZ_X4>(v, i);
}

__device__ __forceinline__ float sum8(float v) {
    v += swzf<SWZ_X1>(v);
    v += swzf<SWZ_X2>(v);
    v += swzf<SWZ_X4>(v);
    return v;
}

// ---------------------------------------------------------------------------
// Init: build zero-padded Wp (16 x DIM) and zero the global accumulators.
// ---------------------------------------------------------------------------
__global__ void moe_init_kernel(const float* __restrict__ W, float* __restrict__ ws) {
    int idx = blockIdx.x * blockDim.x + threadIdx.x;
    const int total = NPAD * DIM;
    for (int i = idx; i < total; i += gridDim.x * blockDim.x) {
        int e = i / DIM;
        ws[WS_WP + i] = (e < NUM_EXPERTS) ? W[i] : 0.0f;
    }
    if (idx < 2 * NUM_EXPERTS) ws[WS_COUNTS + idx] = 0.0f;
}

// ---------------------------------------------------------------------------
// Main router: each wave computes logits for a 16-token tile with
// v_wmma_f32_16x16x4_f32, then softmax/top-2 via swizzle butterflies.
// ---------------------------------------------------------------------------
__global__ __launch_bounds__(256) void moe_router_kernel(
    const float* __restrict__ x, const float* __restrict__ wp,
    float* __restrict__ counts_g, float* __restrict__ psum_g,
    float* __restrict__ out_w, float* __restrict__ out_i) {
    __shared__ float lds_counts[NUM_EXPERTS];
    __shared__ float lds_psum[NUM_EXPERTS];

    const int tid = threadIdx.x;
    if (tid < NUM_EXPERTS) { lds_counts[tid] = 0.0f; lds_psum[tid] = 0.0f; }
    __syncthreads();

    const int wave = tid >> 5;
    const int lane = tid & 31;
    const int half = lane >> 4;     // selects K pair {0,1} vs {2,3}
    const int mrow = lane & 15;     // token row (A) / expert col (B)
    const int tok0 = (blockIdx.x * WAVES_PER_WG + wave) * TOKENS_PER_WAVE;

    // A: lane(M=mrow) holds x[tok][k + 2*half + {0,1}]
    // B: lane(N=mrow) holds Wp[expert][k + 2*half + {0,1}]
    const float* xrow = x  + (size_t)(tok0 + mrow) * DIM + 2 * half;
    const float* wrow = wp + (size_t)mrow * DIM + 2 * half;

    v8f c0 = {};
    v8f c1 = {};
#pragma unroll 4
    for (int k = 0; k < DIM; k += 8) {
        v2f a0 = __builtin_nontemporal_load((const v2f*)(xrow + k));      // stream x, keep Wp cached
        v2f b0 = *(const v2f*)(wrow + k);
        c0 = __builtin_amdgcn_wmma_f32_16x16x4_f32(false, a0, false, b0, (short)0, c0, false, false);
        v2f a1 = __builtin_nontemporal_load((const v2f*)(xrow + k + 4));
        v2f b1 = *(const v2f*)(wrow + k + 4);
        c1 = __builtin_amdgcn_wmma_f32_16x16x4_f32(false, a1, false, b1, (short)0, c1, false, false);
    }
    v8f c = c0 + c1;

    // D layout: VGPR r, lane<16 -> logits[token tok0+r][expert lane];
    //           lane>=16      -> logits[token tok0+r+8][expert lane-16]
    const float NEG_INF = -__builtin_inff();
    float psum_local = 0.0f;
    float keep_w0 = 0.0f, keep_w1 = 0.0f;
    int keep_i1 = 0, keep_i2 = 0;
#pragma unroll
    for (int r = 0; r < 8; ++r) {
        float vv = (mrow < NUM_EXPERTS) ? c[r] : NEG_INF;

        float m1 = vv; int i1 = mrow;
        argmax8(m1, i1);                       // top-1 (ties -> lowest index)
        float vm = (mrow == i1) ? NEG_INF : vv;
        float m2 = vm; int i2 = mrow;
        argmax8(m2, i2);                       // top-2

        float e = __expf(vv - m1);             // padded lanes: -inf - -inf = NaN, never used
        float s = sum8(e);
        psum_local += e / s;                   // this lane's expert prob for token r

        // normalized top-2 softmax pair; token r kept by the lane with mrow==r
        float r2  = __expf(m2 - m1);
        float inv = 1.0f / (1.0f + r2);
        bool mine = (mrow == r);
        keep_w0 = mine ? inv      : keep_w0;
        keep_w1 = mine ? r2 * inv : keep_w1;
        keep_i1 = mine ? i1       : keep_i1;
        keep_i2 = mine ? i2       : keep_i2;
    }

    // coalesced per-token outputs + count atomics from the 8 owner lanes/half
    if (mrow < NUM_EXPERTS) {
        int t = tok0 + mrow + 8 * half;
        v2f wpair = {keep_w0, keep_w1};
        v2f ipair = {(float)keep_i1, (float)keep_i2};
        *(v2f*)(out_w + 2 * t) = wpair;
        *(v2f*)(out_i + 2 * t) = ipair;
        atomicAdd(&lds_counts[keep_i1], 1.0f);
        atomicAdd(&lds_counts[keep_i2], 1.0f);
    }

    // combine the two lane-halves (same expert id), accumulate into LDS
    psum_local += swzf<SWZ_X16>(psum_local);
    if (half == 0 && mrow < NUM_EXPERTS) atomicAdd(&lds_psum[mrow], psum_local);

    __syncthreads();
    if (tid < NUM_EXPERTS) {
        atomicAdd(&counts_g[tid], lds_counts[tid]);
        atomicAdd(&psum_g[tid], lds_psum[tid]);
    }
}

// ---------------------------------------------------------------------------
// Finalize: aux loss + tokens_per_expert
// ---------------------------------------------------------------------------
__global__ void moe_finalize_kernel(const float* __restrict__ counts,
                                    const float* __restrict__ psum,
                                    float* __restrict__ out) {
    int tid = threadIdx.x;
    if (tid < NUM_EXPERTS) out[OUT_TPE + tid] = counts[tid];
    if (tid == 0) {
        float loss = 0.0f;
        for (int e = 0; e < NUM_EXPERTS; ++e) {
            float f = counts[e] / (float)(NUM_TOKENS * TOP_K);
            float p = psum[e] / (float)NUM_TOKENS;
            loss += f * p;
        }
        out[OUT_AUX] = (float)NUM_EXPERTS * loss;
    }
}

extern "C" void kernel_launch(void* const* d_in, const int* in_sizes, int n_in,
                              void* d_out, int out_size, void* d_ws, size_t ws_size,
                              hipStream_t stream) {
    const float* x = (const float*)d_in[0];
    const float* W = (const float*)d_in[1];
    float* out = (float*)d_out;
    float* ws  = (float*)d_ws;

    moe_init_kernel<<<64, 256, 0, stream>>>(W, ws);
    moe_router_kernel<<<NUM_TOKENS / TOKENS_PER_WG, 256, 0, stream>>>(
        x, ws + WS_WP, ws + WS_COUNTS, ws + WS_PSUM, out + OUT_W, out + OUT_I);
    moe_finalize_kernel<<<1, 32, 0, stream>>>(ws + WS_COUNTS, ws + WS_PSUM, out);
}